// LearnableHypergraphConstructor_28174985462030
// MI455X (gfx1250) — compile-verified
//
#include <hip/hip_runtime.h>
#include <math.h>
#include <stdint.h>

// ---------------------------------------------------------------------------
// LearnableHypergraphConstructor pipeline for MI455X (gfx1250, wave32, WMMA)
// All heavy GEMMs use V_WMMA_F32_16X16X4_F32 (f32 in / f32 acc, matrix pipe).
//
// Layout conventions for our intermediates (we control them, so we pick the
// layout the WMMA B-fragment wants): channel-pair interleaved
//     buf[(c>>1)*HW*2 + n*2 + (c&1)]
// so a B-fragment (channels kk,kk+1 at one spatial n, kk always even) is a
// single aligned global_load_b64.
// ---------------------------------------------------------------------------

#define D     256
#define NH    4
#define DH    64
#define TCH   512
#define WINSZ 64
#define BATCH 8
#define HW    4096          // 64*64 spatial
#define C3    768           // 3*D
#define EPSV  1e-5f

#define USE_ASYNC_LDS 1     // CDNA5 global_load_async_to_lds path in k4

typedef __attribute__((ext_vector_type(2))) float v2f;
typedef __attribute__((ext_vector_type(8))) float v8f;

// D = A(16x4) * B(4x16) + C(16x16), f32.  Fragment layouts per CDNA5 ISA 7.12.2:
//   A: lane m = lane&15, K = 2*(lane>>4) + vgpr   (contiguous K pair per lane)
//   B: lane n = lane&15, K = 2*(lane>>4) + vgpr
//   C/D: vgpr r holds M = r + 8*(lane>>4), N = lane&15
__device__ __forceinline__ v8f wmma4(v2f a, v2f b, v8f c) {
    return __builtin_amdgcn_wmma_f32_16x16x4_f32(
        /*neg_a=*/false, a, /*neg_b=*/false, b,
        /*c_mod=*/(short)0, c, /*reuse_a=*/false, /*reuse_b=*/false);
}

__device__ __forceinline__ float gelu_exact(float x) {
    return 0.5f * x * (1.0f + erff(x * 0.70710678118654752f));
}
__device__ __forceinline__ float sigmoid_(float x) {
    return 1.0f / (1.0f + __expf(-x));
}

// ---------------------------------------------------------------------------
// Kernel 0: repack conv2 weights OIHW (o,c,ky,kx) -> (o, tap, c) so WMMA
// A-fragments are contiguous 8-byte loads.
// ---------------------------------------------------------------------------
__global__ void k0_w2t(const float* __restrict__ w2, float* __restrict__ w2t) {
    int idx = blockIdx.x * 256 + threadIdx.x;   // 768*2304 = 1769472
    int o   = idx / 2304;
    int r   = idx - o * 2304;
    int tap = r >> 8;                           // 0..8
    int c   = r & 255;
    w2t[idx] = w2[o * 2304 + c * 9 + tap];
}

// ---------------------------------------------------------------------------
// Kernel 1: 1x1 conv (GEMM 256x4096x256 per batch) + exact GELU.
// One wave per 32(M=outch) x 64(N=spatial) tile: 2 A-frags share each B-frag.
// 4096 waves.  Output X1 is channel-pair interleaved.
// ---------------------------------------------------------------------------
__global__ void k1_conv1_gelu(const float* __restrict__ fm,
                              const float* __restrict__ w1,
                              const float* __restrict__ b1,
                              float* __restrict__ X1) {
    const int lane = threadIdx.x & 31;
    const int wv   = threadIdx.x >> 5;
    const int half = lane >> 4, l15 = lane & 15;
    int gw  = blockIdx.x * 8 + wv;         // 8 * 8 * 64 = 4096 waves
    int b   = gw >> 9;
    int rem = gw & 511;
    int mt  = rem >> 6;                    // 0..7 (32-row d tile)
    int n0  = (rem & 63) << 6;             // spatial base
    const float* xin = fm + b * (D * HW);
    const float* ar0 = w1 + (mt * 32 + l15) * D;
    const float* ar1 = ar0 + 16 * D;
    v8f acc[2][4] = {};
    for (int k0 = 0; k0 < D; k0 += 4) {
        int kk = k0 + half * 2;
        v2f a0 = *(const v2f*)(ar0 + kk);
        v2f a1 = *(const v2f*)(ar1 + kk);
        const float* bp0 = xin + kk * HW + n0 + l15;
        const float* bp1 = bp0 + HW;
#pragma unroll
        for (int j = 0; j < 4; ++j) {
            v2f bb; bb.x = bp0[j * 16]; bb.y = bp1[j * 16];
            acc[0][j] = wmma4(a0, bb, acc[0][j]);
            acc[1][j] = wmma4(a1, bb, acc[1][j]);
        }
    }
    float* xb = X1 + b * (D * HW);
#pragma unroll
    for (int s = 0; s < 2; ++s)
#pragma unroll
        for (int j = 0; j < 4; ++j)
#pragma unroll
            for (int r = 0; r < 8; ++r) {
                int row = mt * 32 + s * 16 + r + 8 * half;
                int n   = n0 + j * 16 + l15;
                xb[((row >> 1) * HW + n) * 2 + (row & 1)] =
                    gelu_exact(acc[s][j][r] + b1[row]);
            }
}

// ---------------------------------------------------------------------------
// Kernel 2: 3x3 conv as implicit GEMM (M=768, N=4096, K=2304 per batch).
// One wave per 32(M) x 64(N = one image row) tile.  y is wave-uniform so the
// row-padding tap skip never diverges; x-edge padding is a clamped load +
// lane select (EXEC stays all-ones around WMMA).  12288 waves, 4608 WMMAs ea.
// A: b64 from repacked weights.  B: b64 from interleaved X1.
// ---------------------------------------------------------------------------
__global__ void k2_conv2(const float* __restrict__ X1,    // interleaved pairs
                         const float* __restrict__ w2t,   // [o][tap][c]
                         const float* __restrict__ b2,
                         float* __restrict__ QKV) {       // interleaved pairs
    const int lane = threadIdx.x & 31;
    const int wv   = threadIdx.x >> 5;
    const int half = lane >> 4, l15 = lane & 15;
    int gw  = blockIdx.x * 8 + wv;         // 8 * 24 * 64 = 12288 waves
    int b   = gw / 1536;
    int rem = gw - b * 1536;
    int mt  = rem >> 6;                    // 0..23 (32-row out-channel tile)
    int y   = rem & 63;                    // image row
    v8f acc[2][4] = {};
    const float* w0  = w2t + (mt * 32 + l15) * (D * 9);
    const float* w1r = w0 + 16 * (D * 9);
    const v2f*   xv  = (const v2f*)(X1 + b * (D * HW));   // pair-indexed
    const v2f    vzero = {};
#pragma unroll
    for (int tap = 0; tap < 9; ++tap) {
        int dy = tap / 3 - 1, dx = tap % 3 - 1;
        int yy = y + dy;
        if (yy < 0 || yy > 63) continue;              // wave-uniform
        const v2f*  xrow = xv + yy * 64;
        // prefetch the start of this tap's input row into near caches
        __builtin_prefetch((const void*)(xrow + l15), 0, 0);
        const float* wt0 = w0  + tap * D;
        const float* wt1 = w1r + tap * D;
        for (int c0 = 0; c0 < D; c0 += 4) {
            int cc = c0 + half * 2;
            v2f a0 = *(const v2f*)(wt0 + cc);
            v2f a1 = *(const v2f*)(wt1 + cc);
            const v2f* bp = xrow + (cc >> 1) * HW;
#pragma unroll
            for (int j = 0; j < 4; ++j) {
                int x   = j * 16 + l15 + dx;
                bool ok = (x >= 0) && (x < 64);
                int xc  = x < 0 ? 0 : (x > 63 ? 63 : x);
                v2f bb  = bp[xc];
                bb = ok ? bb : vzero;
                acc[0][j] = wmma4(a0, bb, acc[0][j]);
                acc[1][j] = wmma4(a1, bb, acc[1][j]);
            }
        }
    }
    float* qb = QKV + b * (C3 * HW);
#pragma unroll
    for (int s = 0; s < 2; ++s)
#pragma unroll
        for (int j = 0; j < 4; ++j)
#pragma unroll
            for (int r = 0; r < 8; ++r) {
                int row = mt * 32 + s * 16 + r + 8 * half;
                int n   = y * 64 + j * 16 + l15;
                qb[((row >> 1) * HW + n) * 2 + (row & 1)] = acc[s][j][r] + b2[row];
            }
}

// ---------------------------------------------------------------------------
// Kernel 3: GroupNorm (4 groups of 192 channels), in place on interleaved QKV.
// channel = g*192 + 2*(i>>13) + (i&1); sums are layout-agnostic.
// ---------------------------------------------------------------------------
__global__ void k3_groupnorm(float* __restrict__ QKV,
                             const float* __restrict__ gw,
                             const float* __restrict__ gb) {
    __shared__ float rs[256], rs2[256];
    int b = blockIdx.x >> 2;
    int g = blockIdx.x & 3;
    const int n = 192 * HW;                // 786432 elems per (b,g)
    float* base = QKV + b * (C3 * HW) + g * n;
    float s = 0.f, s2 = 0.f;
    for (int i = threadIdx.x; i < n; i += 256) {
        float v = base[i];
        s += v; s2 += v * v;
    }
    rs[threadIdx.x] = s; rs2[threadIdx.x] = s2;
    __syncthreads();
    for (int off = 128; off > 0; off >>= 1) {
        if ((int)threadIdx.x < off) {
            rs[threadIdx.x]  += rs[threadIdx.x + off];
            rs2[threadIdx.x] += rs2[threadIdx.x + off];
        }
        __syncthreads();
    }
    float mu   = rs[0] / (float)n;
    float var  = rs2[0] / (float)n - mu * mu;
    float rsig = rsqrtf(var + EPSV);
    for (int i = threadIdx.x; i < n; i += 256) {
        int c = g * 192 + ((i >> 13) << 1) + (i & 1);
        base[i] = (base[i] - mu) * rsig * gw[c] + gb[c];
    }
}

// ---------------------------------------------------------------------------
// Kernel 4: text projection fused with the 16-row mean.
// CDNA5 path: stage the contiguous 16x512 f32 block into LDS with per-lane
// GLOBAL_LOAD_ASYNC_TO_LDS_B128, fence with s_wait_asynccnt, then do the
// column sums out of LDS (the strided reduction pattern LDS is built for).
// ---------------------------------------------------------------------------
__global__ void k4_text(const float* __restrict__ txt,
                        const float* __restrict__ tw,
                        const float* __restrict__ tb,
                        float* __restrict__ TF) {
    __shared__ float sTX[16 * TCH];      // 32 KB staged text rows
    __shared__ float tm[TCH];
    int b = blockIdx.x >> 2;
    int i = blockIdx.x & 3;
    const float* src = txt + (b * 64 + i * 16) * TCH;   // contiguous 8192 f32
#if USE_ASYNC_LDS
#pragma unroll
    for (int it = 0; it < 8; ++it) {
        int f = (threadIdx.x + it * 256) * 4;           // 4 floats per b128
        unsigned lds_off = (unsigned)(uintptr_t)(sTX + f);
        unsigned long long ga = (unsigned long long)(uintptr_t)(src + f);
        asm volatile("global_load_async_to_lds_b128 %0, %1, off"
                     :: "v"(lds_off), "v"(ga)
                     : "memory");
    }
    asm volatile("s_wait_asynccnt 0x0" ::: "memory");
#else
    for (int f = threadIdx.x; f < 16 * TCH; f += 256) sTX[f] = src[f];
#endif
    __syncthreads();
    for (int tc = threadIdx.x; tc < TCH; tc += 256) {
        float s = 0.f;
        for (int l = 0; l < 16; ++l) s += sTX[l * TCH + tc];
        tm[tc] = s * (1.0f / 16.0f);
    }
    __syncthreads();
    int d = threadIdx.x;
    float acc = tb[d];
    const float* wr = tw + d * TCH;
    for (int tc = 0; tc < TCH; ++tc) acc += wr[tc] * tm[tc];
    TF[(b * 4 + i) * D + d] = acc;
}

// ---------------------------------------------------------------------------
// Kernel 5: local windowed MHA.  One 8-wave block per window (512 windows).
// Matrices kept dim-major (matching interleaved qkv memory) -> no transposes.
// LDS rows padded to 65 floats (65 mod 64 = 1 -> conflict-free columns).
// ---------------------------------------------------------------------------
__global__ void k5_local_mha(const float* __restrict__ QKV,   // interleaved
                             const float* __restrict__ ipw,
                             const float* __restrict__ ipb,
                             const float* __restrict__ opw,
                             const float* __restrict__ opb,
                             float* __restrict__ LOC) {       // interleaved
    extern __shared__ float smem[];
    float* sQ = smem;                 // 64 x 65 (dim, token), pre-scaled 1/8
    float* sK = sQ + 64 * 65;
    float* sV = sK + 64 * 65;
    float* sS = sV + 64 * 65;         // (qt, kt) scores / attn
    float* sO = sS + 64 * 65;         // 256 x 65 (dim, token) concat heads
    const int lane = threadIdx.x & 31;
    const int wv   = threadIdx.x >> 5;
    const int half = lane >> 4, l15 = lane & 15;
    int b  = blockIdx.x >> 6;
    int w  = blockIdx.x & 63;
    int n0 = w * 64;

    for (int h = 0; h < NH; ++h) {
        // -- phase 1: head projections Q/K/V (48 16x16 tiles over 8 waves) --
#pragma unroll 1
        for (int i = 0; i < 6; ++i) {
            int tix = wv * 6 + i;
            int mat = tix >> 4;                  // 0=q 1=k 2=v
            int rm  = tix & 15;
            int mt = rm >> 2, nt = rm & 3;
            int wrow0 = mat * D + h * DH + mt * 16;
            const float* arow = ipw + (wrow0 + l15) * D;
            const v2f*   xin  = (const v2f*)(QKV + b * (C3 * HW) + mat * (D * HW));
            int col = n0 + nt * 16 + l15;
            v8f acc = {};
            for (int k0 = 0; k0 < D; k0 += 4) {
                int kk = k0 + half * 2;
                v2f a  = *(const v2f*)(arow + kk);
                v2f bb = xin[(kk >> 1) * HW + col];
                acc = wmma4(a, bb, acc);
            }
            float scale = (mat == 0) ? 0.125f : 1.0f;   // fold 1/sqrt(dh)
            float* dst = (mat == 0) ? sQ : (mat == 1 ? sK : sV);
#pragma unroll
            for (int r = 0; r < 8; ++r) {
                int m = mt * 16 + r + 8 * half;
                dst[m * 65 + nt * 16 + l15] =
                    (acc[r] + ipb[wrow0 + r + 8 * half]) * scale;
            }
        }
        __syncthreads();
        // -- phase 2: S = Q^T K (16 tiles over 8 waves) --
#pragma unroll 1
        for (int i = 0; i < 2; ++i) {
            int tix = wv * 2 + i;
            int mt = tix >> 2, nt = tix & 3;
            v8f acc = {};
            for (int k0 = 0; k0 < DH; k0 += 4) {
                int kk = k0 + half * 2;
                v2f a; a.x = sQ[kk * 65 + mt * 16 + l15];
                       a.y = sQ[(kk + 1) * 65 + mt * 16 + l15];
                v2f bb; bb.x = sK[kk * 65 + nt * 16 + l15];
                        bb.y = sK[(kk + 1) * 65 + nt * 16 + l15];
                acc = wmma4(a, bb, acc);
            }
#pragma unroll
            for (int r = 0; r < 8; ++r)
                sS[(mt * 16 + r + 8 * half) * 65 + nt * 16 + l15] = acc[r];
        }
        __syncthreads();
        // -- phase 3: softmax over rows (64 rows, one thread each) --
        if (threadIdx.x < 64) {
            float* row = sS + threadIdx.x * 65;
            float m = row[0];
            for (int k = 1; k < 64; ++k) m = fmaxf(m, row[k]);
            float s = 0.f;
            for (int k = 0; k < 64; ++k) { float e = __expf(row[k] - m); row[k] = e; s += e; }
            float inv = 1.0f / s;
            for (int k = 0; k < 64; ++k) row[k] *= inv;
        }
        __syncthreads();
        // -- phase 4: O_h = V * attn^T  into sO rows [h*64, h*64+64) --
#pragma unroll 1
        for (int i = 0; i < 2; ++i) {
            int tix = wv * 2 + i;
            int mt = tix >> 2, nt = tix & 3;
            v8f acc = {};
            for (int k0 = 0; k0 < WINSZ; k0 += 4) {
                int kk = k0 + half * 2;
                v2f a; a.x = sV[(mt * 16 + l15) * 65 + kk];
                       a.y = sV[(mt * 16 + l15) * 65 + kk + 1];
                v2f bb; bb.x = sS[(nt * 16 + l15) * 65 + kk];
                        bb.y = sS[(nt * 16 + l15) * 65 + kk + 1];
                acc = wmma4(a, bb, acc);
            }
#pragma unroll
            for (int r = 0; r < 8; ++r)
                sO[(h * DH + mt * 16 + r + 8 * half) * 65 + nt * 16 + l15] = acc[r];
        }
        __syncthreads();
    }
    // -- phase 5: output projection, store interleaved local (B,256,4096) --
#pragma unroll 1
    for (int i = 0; i < 8; ++i) {
        int tix = wv * 8 + i;
        int mt = tix >> 2, nt = tix & 3;
        const float* arow = opw + (mt * 16 + l15) * D;
        v8f acc = {};
        for (int k0 = 0; k0 < D; k0 += 4) {
            int kk = k0 + half * 2;
            v2f a = *(const v2f*)(arow + kk);
            v2f bb; bb.x = sO[kk * 65 + nt * 16 + l15];
                    bb.y = sO[(kk + 1) * 65 + nt * 16 + l15];
            acc = wmma4(a, bb, acc);
        }
        float* lb = LOC + b * (D * HW);
#pragma unroll
        for (int r = 0; r < 8; ++r) {
            int row = mt * 16 + r + 8 * half;
            int n   = n0 + nt * 16 + l15;
            lb[((row >> 1) * HW + n) * 2 + (row & 1)] = acc[r] + opb[row];
        }
    }
}

// ---------------------------------------------------------------------------
// Kernel 5b: global_q[b][w][o] = mean_t local[b][o][w*64+t]  (interleaved LOC)
// ---------------------------------------------------------------------------
__global__ void k5b_gqmean(const float* __restrict__ LOC, float* __restrict__ GQ) {
    int idx = blockIdx.x * 256 + threadIdx.x;    // 8*64*256 = 131072
    int b   = idx >> 14;
    int rem = idx & 16383;
    int w   = rem >> 8;
    int o   = rem & 255;
    const float* p = LOC + b * (D * HW) + ((o >> 1) * HW + w * 64) * 2 + (o & 1);
    float s = 0.f;
    for (int t = 0; t < 64; ++t) s += p[2 * t];
    GQ[(b * 64 + w) * D + o] = s * (1.0f / 64.0f);
}

// ---------------------------------------------------------------------------
// Kernel 6: global MHA against 4 text tokens (tiny -> plain VALU).
// One block per batch.  Writes glob dim-major (B,256,64), plain layout.
// ---------------------------------------------------------------------------
__global__ void k6_global_mha(const float* __restrict__ TF,
                              const float* __restrict__ GQ,
                              const float* __restrict__ ipw,
                              const float* __restrict__ ipb,
                              const float* __restrict__ opw,
                              const float* __restrict__ opb,
                              float* __restrict__ GLOB) {
    extern __shared__ float smem[];
    float* tf = smem;            // 4 x 256
    float* kg = tf + 1024;       // 4 x 256
    float* vg = kg + 1024;       // 4 x 256
    float* qg = vg + 1024;       // 64 x 256
    float* oh = qg + 16384;      // 64 x 256
    int b = blockIdx.x;
    int tid = threadIdx.x;
    for (int j = 0; j < 4; ++j) tf[j * 256 + tid] = TF[b * 1024 + j * 256 + tid];
    __syncthreads();
    {   // project K,V for the 4 text tokens
        int o = tid;
        const float* wk  = ipw + (D + o) * D;
        const float* wvp = ipw + (2 * D + o) * D;
        for (int j = 0; j < 4; ++j) {
            float ak = ipb[D + o], av = ipb[2 * D + o];
            const float* t = tf + j * 256;
            for (int d = 0; d < D; ++d) { ak += wk[d] * t[d]; av += wvp[d] * t[d]; }
            kg[j * 256 + o] = ak; vg[j * 256 + o] = av;
        }
    }
    for (int it = 0; it < 64; ++it) {            // project 64 queries
        int idx = tid + it * 256;
        int t = idx >> 8, o = idx & 255;
        float acc = ipb[o];
        const float* wq = ipw + o * D;
        const float* q  = GQ + (b * 64 + t) * D;
        for (int d = 0; d < D; ++d) acc += wq[d] * q[d];
        qg[t * 256 + o] = acc;
    }
    __syncthreads();
    {   // attention: thread -> (query t, head h), 4 keys
        int t = tid >> 2, h = tid & 3;
        float* qrow = qg + t * 256 + h * 64;
        float s[4];
        float m = -1e30f;
        for (int j = 0; j < 4; ++j) {
            float a = 0.f;
            const float* kr = kg + j * 256 + h * 64;
            for (int dd = 0; dd < 64; ++dd) a += qrow[dd] * kr[dd];
            s[j] = a * 0.125f;
            m = fmaxf(m, s[j]);
        }
        float sum = 0.f;
        for (int j = 0; j < 4; ++j) { s[j] = __expf(s[j] - m); sum += s[j]; }
        float inv = 1.0f / sum;
        for (int j = 0; j < 4; ++j) s[j] *= inv;
        float* orow = oh + t * 256 + h * 64;
        for (int dd = 0; dd < 64; ++dd) {
            float a = 0.f;
            for (int j = 0; j < 4; ++j) a += s[j] * vg[j * 256 + h * 64 + dd];
            orow[dd] = a;
        }
    }
    __syncthreads();
    for (int it = 0; it < 64; ++it) {            // output projection
        int idx = tid + it * 256;
        int o = idx >> 6, t = idx & 63;
        float acc = opb[o];
        const float* wo = opw + o * D;
        const float* orow = oh + t * 256;
        for (int d = 0; d < D; ++d) acc += wo[d] * orow[d];
        GLOB[b * (D * WINSZ) + o * WINSZ + t] = acc;
    }
}

// ---------------------------------------------------------------------------
// Kernel 7: fused MLP epilogue per window.
// h = gelu([local | glob] @ ep1^T + b1) (WMMA, K=512),
// out = sigmoid(h @ ep2^T + b2)        (WMMA, K=256).
// local part of B is one b64 from interleaved LOC; glob part is a broadcast.
// ---------------------------------------------------------------------------
__global__ void k7_epilogue(const float* __restrict__ LOC,    // interleaved
                            const float* __restrict__ GLOB,
                            const float* __restrict__ e1w,
                            const float* __restrict__ e1b,
                            const float* __restrict__ e2w,
                            const float* __restrict__ e2b,
                            float* __restrict__ out) {
    extern __shared__ float smem[];
    float* sH = smem;                 // 256 x 65 (e, token)
    const int lane = threadIdx.x & 31;
    const int wv   = threadIdx.x >> 5;
    const int half = lane >> 4, l15 = lane & 15;
    int b = blockIdx.x >> 6;
    int w = blockIdx.x & 63;
    const v2f*   locv = (const v2f*)(LOC + b * (D * HW));
    const float* glb  = GLOB + b * (D * WINSZ) + w;     // column w, broadcast
#pragma unroll 1
    for (int i = 0; i < 8; ++i) {
        int tix = wv * 8 + i;
        int mt = tix >> 2, nt = tix & 3;
        const float* arow = e1w + (mt * 16 + l15) * (2 * D);
        int col = w * 64 + nt * 16 + l15;
        v8f acc = {};
        for (int k0 = 0; k0 < 2 * D; k0 += 4) {
            int kk = k0 + half * 2;          // even; region uniform per wave
            v2f a = *(const v2f*)(arow + kk);
            v2f bb;
            if (kk < D) {
                bb = locv[(kk >> 1) * HW + col];
            } else {
                bb.x = glb[(kk - D) * WINSZ];
                bb.y = glb[(kk + 1 - D) * WINSZ];
            }
            acc = wmma4(a, bb, acc);
        }
#pragma unroll
        for (int r = 0; r < 8; ++r) {
            int row = mt * 16 + r + 8 * half;
            sH[row * 65 + nt * 16 + l15] = gelu_exact(acc[r] + e1b[row]);
        }
    }
    __syncthreads();
    if (wv < 4) {                       // M=16, N=64: one 16x16 tile per wave
        int nt = wv;
        const float* arow = e2w + l15 * D;
        v8f acc = {};
        for (int k0 = 0; k0 < D; k0 += 4) {
            int kk = k0 + half * 2;
            v2f a = *(const v2f*)(arow + kk);
            v2f bb; bb.x = sH[kk * 65 + nt * 16 + l15];
                    bb.y = sH[(kk + 1) * 65 + nt * 16 + l15];
            acc = wmma4(a, bb, acc);
        }
#pragma unroll
        for (int r = 0; r < 8; ++r) {
            int ne = r + 8 * half;
            int n  = w * 64 + nt * 16 + l15;
            out[(b * HW + n) * 16 + ne] = sigmoid_(acc[r] + e2b[ne]);
        }
    }
}

// ---------------------------------------------------------------------------
extern "C" void kernel_launch(void* const* d_in, const int* in_sizes, int n_in,
                              void* d_out, int out_size, void* d_ws, size_t ws_size,
                              hipStream_t stream) {
    const float* fm  = (const float*)d_in[0];
    const float* txt = (const float*)d_in[1];
    const float* c1w = (const float*)d_in[2];
    const float* c1b = (const float*)d_in[3];
    const float* c2w = (const float*)d_in[4];
    const float* c2b = (const float*)d_in[5];
    const float* gnw = (const float*)d_in[6];
    const float* gnb = (const float*)d_in[7];
    const float* ipw = (const float*)d_in[8];
    const float* ipb = (const float*)d_in[9];
    const float* opw = (const float*)d_in[10];
    const float* opb = (const float*)d_in[11];
    const float* tw  = (const float*)d_in[12];
    const float* tb  = (const float*)d_in[13];
    const float* e1w = (const float*)d_in[14];
    const float* e1b = (const float*)d_in[15];
    const float* e2w = (const float*)d_in[16];
    const float* e2b = (const float*)d_in[17];
    float* out = (float*)d_out;

    float* ws   = (float*)d_ws;
    float* X1   = ws;                        // 8*256*4096  (32 MiB, interleaved)
    float* QKV  = X1 + BATCH * D * HW;       // 8*768*4096  (96 MiB, interleaved)
    float* TFb  = QKV + BATCH * C3 * HW;     // 8*4*256
    float* GQ   = TFb + BATCH * 4 * D;       // 8*64*256
    float* GLOB = GQ + BATCH * 64 * D;       // 8*256*64
    float* W2T  = GLOB + BATCH * D * WINSZ;  // 768*2304 repacked conv2 weights
    float* LOC  = X1;                        // reuse X1 region after conv2

    k0_w2t       <<<6912, 256, 0, stream>>>(c2w, W2T);
    k1_conv1_gelu<<<512,  256, 0, stream>>>(fm, c1w, c1b, X1);
    k2_conv2     <<<1536, 256, 0, stream>>>(X1, W2T, c2b, QKV);
    k3_groupnorm <<<32,   256, 0, stream>>>(QKV, gnw, gnb);
    k4_text      <<<32,   256, 0, stream>>>(txt, tw, tb, TFb);
    k5_local_mha <<<512,  256, (4 * 64 * 65 + 256 * 65) * sizeof(float), stream>>>(
        QKV, ipw, ipb, opw, opb, LOC);
    k5b_gqmean   <<<512,  256, 0, stream>>>(LOC, GQ);
    k6_global_mha<<<8,    256, (3 * 1024 + 2 * 16384) * sizeof(float), stream>>>(
        TFb, GQ, ipw, ipb, opw, opb, GLOB);
    k7_epilogue  <<<512,  256, 256 * 65 * sizeof(float), stream>>>(
        LOC, GLOB, e1w, e1b, e2w, e2b, out);
}